// AdaptiveAvgPoolSequence_6554120094033
// MI455X (gfx1250) — compile-verified
//
#include <hip/hip_runtime.h>
#include <cstdint>

#define HGRID   16
#define KBINS   256              // 16*16 bins
#define BATCH   4
#define CHAN    64
#define CH_HALF 128              // channels handled per workgroup (blockIdx.y splits)
#define CHUNK   1024             // points staged per iteration
#define WG_THREADS 1024

// Issue one per-lane async global->LDS copy of a coord pair (8 bytes).
__device__ __forceinline__ void stage_coords_async(const float* __restrict__ coords,
                                                   int gi, int N,
                                                   float* dstLds) {
  if (gi < N) {
    unsigned laddr = (unsigned)(uintptr_t)dstLds;
    const float* gsrc = coords + 2 * (size_t)gi;
    asm volatile("global_load_async_to_lds_b64 %0, %1, off"
                 :: "v"(laddr), "v"(gsrc) : "memory");
  }
}

__global__ void __launch_bounds__(WG_THREADS, 1)
pool_segsum_kernel(const float* __restrict__ coords,
                   const float* __restrict__ values,
                   float* __restrict__ out,
                   float* __restrict__ counts,
                   int N, int nChunks) {
  // 128 KB accumulator + 2x8 KB coords stage + 4 KB bins + 1 KB counts ≈ 149 KB LDS
  __shared__ float lds_acc[KBINS * CH_HALF];
  __shared__ float lds_coords[2][CHUNK * 2];
  __shared__ int   lds_bins[CHUNK];
  __shared__ float lds_counts[KBINS];

  const int tid  = threadIdx.x;
  const int half = blockIdx.y;               // 0 or 1: which 128 channels
  const int chl  = tid & (CH_HALF - 1);      // local channel 0..127
  const int ch   = chl + half * CH_HALF;     // global combined channel 0..255
  const int b    = ch >> 6;                  // batch 0..3
  const int c    = ch & 63;                  // channel-in-batch
  const int grp  = tid >> 7;                 // 8 point-groups of 128 threads

  const float* __restrict__ pv =
      values + (size_t)b * (size_t)N * CHAN + (size_t)c;

  for (int i = tid; i < KBINS * CH_HALF; i += WG_THREADS) lds_acc[i] = 0.0f;
  if (tid < KBINS) lds_counts[tid] = 0.0f;
  __syncthreads();

  // bin edges: linspace(-1-eps, 1+eps, 17); bin = floor((x - lo)/step)
  const float kOff   = 1.0f + 1.0e-6f;
  const float kScale = (float)HGRID / (2.0f + 2.0e-6f);

  // Preload first chunk's coords into buffer 0 (double-buffered async staging).
  stage_coords_async(coords, blockIdx.x * CHUNK + tid, N, &lds_coords[0][2 * tid]);

  int it = 0;
  for (int chunk = blockIdx.x; chunk < nChunks; chunk += gridDim.x, ++it) {
    const int base = chunk * CHUNK;
    const int buf  = it & 1;
    const int nextChunk = chunk + gridDim.x;

    // ---- overlap: kick off next chunk's coords, then wait for current ----
    if (nextChunk < nChunks) {
      stage_coords_async(coords, nextChunk * CHUNK + tid, N,
                         &lds_coords[buf ^ 1][2 * tid]);
      asm volatile("s_wait_asynccnt 1" ::: "memory");  // current chunk landed
    } else {
      asm volatile("s_wait_asynccnt 0" ::: "memory");
    }

    // ---- stage A: compute bins from staged coords ----
    const int gi = base + tid;
    if (gi < N) {
      float x = lds_coords[buf][2 * tid];
      float y = lds_coords[buf][2 * tid + 1];
      int bx = (int)((x + kOff) * kScale);
      int by = (int)((y + kOff) * kScale);
      bx = bx < 0 ? 0 : (bx > HGRID - 1 ? HGRID - 1 : bx);
      by = by < 0 ? 0 : (by > HGRID - 1 ? HGRID - 1 : by);
      int bin = bx + (by << 4);
      lds_bins[tid] = bin;
      if (half == 0) atomicAdd(&lds_counts[bin], 1.0f);   // ds_add_f32
    } else {
      lds_bins[tid] = -1;
    }
    __syncthreads();

    // prefetch a slice of the next chunk's value stream
    if (nextChunk < nChunks) {
      size_t pn = (size_t)(nextChunk * CHUNK + grp * (CHUNK / 8));
      __builtin_prefetch(pv + pn * CHAN, 0, 1);
    }

    // ---- stage B: stream values, scatter-add into LDS-privatized bins ----
    const int pbase = grp * (CHUNK / 8);     // 128 points per group
    if (base + CHUNK <= N) {
      // Hot path: full chunk, no guards in the inner loop.
      for (int p0 = 0; p0 < CHUNK / 8; p0 += 8) {
        int   bn[8];
        float v[8];
#pragma unroll
        for (int u = 0; u < 8; ++u) {
          int p = pbase + p0 + u;
          bn[u] = lds_bins[p];                       // ds_load (merges to b128)
          v[u]  = pv[(size_t)(base + p) * CHAN];     // coalesced global_load_b32
        }
#pragma unroll
        for (int u = 0; u < 8; ++u)
          atomicAdd(&lds_acc[(bn[u] << 7) + chl], v[u]);  // ds_add_f32
      }
    } else {
      // Tail chunk: guarded.
      for (int p0 = 0; p0 < CHUNK / 8; p0 += 8) {
        int   bn[8];
        float v[8];
#pragma unroll
        for (int u = 0; u < 8; ++u) {
          int p = pbase + p0 + u;
          bn[u] = lds_bins[p];
          size_t n = (size_t)(base + p);
          if (n >= (size_t)N) n = (size_t)N - 1;
          v[u] = pv[n * CHAN];
        }
#pragma unroll
        for (int u = 0; u < 8; ++u)
          if (bn[u] >= 0)
            atomicAdd(&lds_acc[(bn[u] << 7) + chl], v[u]);
      }
    }
    __syncthreads();
  }

  // ---- flush private accumulator to global with native f32 atomics ----
  for (int j = 0; j < (KBINS * CH_HALF) / WG_THREADS; ++j) {
    int idx = j * WG_THREADS + tid;
    int k   = idx >> 7;                 // bin
    int cl  = idx & (CH_HALF - 1);
    int cc  = cl + half * CH_HALF;      // combined channel
    int bb  = cc >> 6;
    int c2  = cc & 63;
    unsafeAtomicAdd(&out[(size_t)bb * (KBINS * CHAN) + (size_t)k * CHAN + c2],
                    lds_acc[idx]);      // global_atomic_add_f32
  }
  if (half == 0 && tid < KBINS)
    unsafeAtomicAdd(&counts[tid], lds_counts[tid]);
}

__global__ void __launch_bounds__(256)
pool_zero_kernel(float* __restrict__ out, float* __restrict__ counts) {
  int idx = blockIdx.x * 256 + threadIdx.x;
  if (idx < BATCH * KBINS * CHAN) out[idx] = 0.0f;
  if (idx < KBINS) counts[idx] = 0.0f;
}

__global__ void __launch_bounds__(256)
pool_divide_kernel(float* __restrict__ out, const float* __restrict__ counts) {
  int idx = blockIdx.x * 256 + threadIdx.x;
  if (idx >= BATCH * KBINS * CHAN) return;
  int k = (idx >> 6) & (KBINS - 1);     // out layout: b*16384 + k*64 + c
  float cnt = counts[k];
  out[idx] = cnt > 0.0f ? out[idx] / cnt : 0.0f;
}

extern "C" void kernel_launch(void* const* d_in, const int* in_sizes, int n_in,
                              void* d_out, int out_size, void* d_ws, size_t ws_size,
                              hipStream_t stream) {
  const float* coords = (const float*)d_in[0];   // [N,2] f32
  const float* values = (const float*)d_in[1];   // [B,N,C] f32
  float* out    = (float*)d_out;                 // [B, K*C] f32 (sums, then means)
  float* counts = (float*)d_ws;                  // [K] f32 scratch

  const int N = in_sizes[0] / 2;
  const int total = BATCH * KBINS * CHAN;

  pool_zero_kernel<<<(total + 255) / 256, 256, 0, stream>>>(out, counts);

  const int nChunks = (N + CHUNK - 1) / CHUNK;
  int gx = nChunks < 256 ? (nChunks > 0 ? nChunks : 1) : 256;
  dim3 grid(gx, 2);
  pool_segsum_kernel<<<grid, WG_THREADS, 0, stream>>>(coords, values, out,
                                                      counts, N, nChunks);

  pool_divide_kernel<<<(total + 255) / 256, 256, 0, stream>>>(out, counts);
}